// IRPropPlusBackprop_33844342292853
// MI455X (gfx1250) — compile-verified
//
#include <hip/hip_runtime.h>

// iRprop+ elementwise optimizer update for MI455X (gfx1250).
// Bandwidth-bound: 36 B/element fp32 streaming -> optimize for wide NT VMEM.

typedef __attribute__((ext_vector_type(4))) float f4;

#define STEP_MIN_C  1e-6f
#define STEP_MAX_C  50.0f
#define ETA_PLUS_C  1.2f
#define ETA_MINUS_C 0.5f

__device__ __forceinline__ void irprop_scalar(
    float p, float g, float s, float pg, float pu, bool loss_inc,
    float& np, float& ns, float& npg, float& nu)
{
    const float sign_agree = g * pg;
    const float s_up = fminf(s * ETA_PLUS_C,  STEP_MAX_C);
    const float s_dn = fmaxf(s * ETA_MINUS_C, STEP_MIN_C);
    const float snew = (sign_agree > 0.0f) ? s_up
                     : ((sign_agree < 0.0f) ? s_dn : s);
    const bool  neg  = (sign_agree < 0.0f);

    // jnp.sign(g): +1 / -1 / 0
    const float sgn_g = (g > 0.0f) ? 1.0f : ((g < 0.0f) ? -1.0f : 0.0f);

    float upd = -snew * sgn_g;
    const float bt = loss_inc ? -pu : 0.0f;
    upd = neg ? bt : upd;

    np  = __fadd_rn(p, upd);   // unfused add: keep bit-consistent with stored upd
    ns  = snew;
    npg = neg ? 0.0f : g;
    nu  = upd;
}

__global__ __launch_bounds__(256) void irprop_vec4_kernel(
    const f4* __restrict__ param,
    const f4* __restrict__ grad,
    const f4* __restrict__ step,
    const f4* __restrict__ pgrad,
    const f4* __restrict__ pupd,
    const float* __restrict__ loss_p,
    const float* __restrict__ ploss_p,
    f4* __restrict__ out_param,
    f4* __restrict__ out_step,
    f4* __restrict__ out_pgrad,
    f4* __restrict__ out_upd,
    int n4)
{
    const int i = blockIdx.x * blockDim.x + threadIdx.x;
    if (i >= n4) return;

    // Wave-uniform scalar condition (lifts to s_load + SALU compare).
    const bool loss_inc = loss_p[0] > ploss_p[0];

    // Streaming 128-bit non-temporal loads (global_load_b128, TH=NT).
    const f4 p  = __builtin_nontemporal_load(&param[i]);
    const f4 g  = __builtin_nontemporal_load(&grad[i]);
    const f4 s  = __builtin_nontemporal_load(&step[i]);
    const f4 pg = __builtin_nontemporal_load(&pgrad[i]);
    const f4 pu = __builtin_nontemporal_load(&pupd[i]);

    f4 np, ns, npg, nu;
#pragma unroll
    for (int k = 0; k < 4; ++k) {
        float a, b, c, d;
        irprop_scalar(p[k], g[k], s[k], pg[k], pu[k], loss_inc, a, b, c, d);
        np[k] = a; ns[k] = b; npg[k] = c; nu[k] = d;
    }

    // Streaming 128-bit non-temporal stores (global_store_b128, TH=NT).
    __builtin_nontemporal_store(np,  &out_param[i]);
    __builtin_nontemporal_store(ns,  &out_step[i]);
    __builtin_nontemporal_store(npg, &out_pgrad[i]);
    __builtin_nontemporal_store(nu,  &out_upd[i]);
}

// Robustness tail for n % 4 != 0 (not expected for N = 2^25).
__global__ __launch_bounds__(64) void irprop_tail_kernel(
    const float* __restrict__ param,
    const float* __restrict__ grad,
    const float* __restrict__ step,
    const float* __restrict__ pgrad,
    const float* __restrict__ pupd,
    const float* __restrict__ loss_p,
    const float* __restrict__ ploss_p,
    float* __restrict__ out_param,
    float* __restrict__ out_step,
    float* __restrict__ out_pgrad,
    float* __restrict__ out_upd,
    int start, int n)
{
    const int i = start + blockIdx.x * blockDim.x + threadIdx.x;
    if (i >= n) return;
    const bool loss_inc = loss_p[0] > ploss_p[0];
    float a, b, c, d;
    irprop_scalar(param[i], grad[i], step[i], pgrad[i], pupd[i], loss_inc, a, b, c, d);
    out_param[i] = a;
    out_step[i]  = b;
    out_pgrad[i] = c;
    out_upd[i]   = d;
}

extern "C" void kernel_launch(void* const* d_in, const int* in_sizes, int n_in,
                              void* d_out, int out_size, void* d_ws, size_t ws_size,
                              hipStream_t stream) {
    (void)n_in; (void)d_ws; (void)ws_size; (void)out_size;

    const float* param = (const float*)d_in[0];
    const float* grad  = (const float*)d_in[1];
    const float* step  = (const float*)d_in[2];
    const float* pgrad = (const float*)d_in[3];
    const float* pupd  = (const float*)d_in[4];
    const float* loss  = (const float*)d_in[5];
    const float* ploss = (const float*)d_in[6];

    const int n  = in_sizes[0];
    const int n4 = n >> 2;           // 128-bit vector groups
    const int tail_start = n4 << 2;

    float* out       = (float*)d_out;
    float* out_param = out;
    float* out_step  = out + (size_t)n;
    float* out_pgrad = out + 2 * (size_t)n;
    float* out_upd   = out + 3 * (size_t)n;

    if (n4 > 0) {
        const int threads = 256;
        const int blocks  = (n4 + threads - 1) / threads;
        irprop_vec4_kernel<<<blocks, threads, 0, stream>>>(
            (const f4*)param, (const f4*)grad, (const f4*)step,
            (const f4*)pgrad, (const f4*)pupd, loss, ploss,
            (f4*)out_param, (f4*)out_step, (f4*)out_pgrad, (f4*)out_upd, n4);
    }
    if (tail_start < n) {
        const int rem = n - tail_start;
        irprop_tail_kernel<<<(rem + 63) / 64, 64, 0, stream>>>(
            param, grad, step, pgrad, pupd, loss, ploss,
            out_param, out_step, out_pgrad, out_upd, tail_start, n);
    }
}